// PiNet_5858335392264
// MI455X (gfx1250) — compile-verified
//
#include <hip/hip_runtime.h>
#include <hip/hip_bf16.h>
#include <stddef.h>

// ---------------------------------------------------------------------------
// PiNet VQ-VAE forward for MI455X (gfx1250), fp32 WMMA (V_WMMA_F32_16X16X4_F32)
// Block tile 128(M) x 128(N), 8 waves, 8 N-tiles per wave, K staged via LDS.
// Tile staging uses GLOBAL_LOAD_ASYNC_TO_LDS_B128 (ASYNCcnt path) when the
// toolchain declares the builtin, else falls back to float4 load/store.
// ---------------------------------------------------------------------------

typedef __attribute__((ext_vector_type(2))) float v2f;
typedef __attribute__((ext_vector_type(8))) float v8f;
typedef __attribute__((ext_vector_type(4))) int   v4i;

#define B_ROWS   8192
#define PROP_DIM 128
#define FUT_DIM  256
#define Z_LEN    64
#define CODE_DIM 16
#define NUM_EMB  4096
#define ACT_DIM  32

#define XS_STRIDE 20    // 16 cols + pad: conflict-free A-frag b64 reads, 16B aligned
#define WS_STRIDE 136   // 128 cols + pad: 2*S mod 64 = 16 -> no cross-half conflicts

#if defined(__gfx1250__) && \
    __has_builtin(__builtin_amdgcn_global_load_async_to_lds_b128) && \
    __has_builtin(__builtin_amdgcn_s_wait_asynccnt)
#define USE_ASYNC_LDS 1
typedef __attribute__((address_space(1))) v4i as1_v4i;   // global <4 x i32>
typedef __attribute__((address_space(3))) v4i as3_v4i;   // LDS    <4 x i32>
// addrspacecast-based helper (generic -> AS1 / AS3); const dropped explicitly
__device__ __forceinline__ void async_copy_b128(const float* gp, float* lp)
{
    float* g = const_cast<float*>(gp);
    __builtin_amdgcn_global_load_async_to_lds_b128(
        (as1_v4i*)g, (as3_v4i*)lp, 0, 0);
}
#else
#define USE_ASYNC_LDS 0
#endif

// ---------------------------------------------------------------------------
// Generic GEMM:  Y[M,N] = act( X[M,K] @ W[K,N] + bias[N] )
// block = 256 threads (8 waves). Block tile 128(M) x 128(N). K chunked by 16.
// Each wave owns a 16(M) x 128(N) strip = 8 accumulators; per K-chunk it
// issues 4 ks-steps x 8 n-tiles = 32 v_wmma_f32_16x16x4_f32.
// Requires: M % 128 == 0, K % 16 == 0, N % 4 == 0. N bounds-checked.
// ---------------------------------------------------------------------------
__global__ __launch_bounds__(256)
void gemm_bias_act(const float* __restrict__ X, const float* __restrict__ W,
                   const float* __restrict__ bias, float* __restrict__ Y,
                   int M, int K, int N, int relu)
{
    __shared__ float Xs[128 * XS_STRIDE];
    __shared__ float Ws[16 * WS_STRIDE];

    const int tid  = threadIdx.x;
    const int wv   = tid >> 5;
    const int lane = tid & 31;
    const int half = lane >> 4;   // 0 or 1
    const int l16  = lane & 15;

    const int bm = blockIdx.y * 128;
    const int bn = blockIdx.x * 128;

    v8f acc[8] = {};   // zero-initialized accumulators

    for (int kc = 0; kc < K; kc += 16) {
        // ---- stage X tile: 128 rows x 16 cols (512 float4 chunks) ----
        #pragma unroll
        for (int idx = tid; idx < 512; idx += 256) {
            const int r  = idx >> 2;
            const int c0 = (idx & 3) << 2;
            const float* gp = X + (size_t)(bm + r) * K + kc + c0;
            float* lp = &Xs[r * XS_STRIDE + c0];
#if USE_ASYNC_LDS
            async_copy_b128(gp, lp);
#else
            *(float4*)lp = *(const float4*)gp;
#endif
        }
        // ---- stage W tile: 16 rows x 128 cols (512 float4 chunks) ----
        #pragma unroll
        for (int idx = tid; idx < 512; idx += 256) {
            const int r  = idx >> 5;          // 0..15
            const int c0 = (idx & 31) << 2;   // 0..124
            float* lp = &Ws[r * WS_STRIDE + c0];
            if (bn + c0 < N) {
                const float* gp = W + (size_t)(kc + r) * N + bn + c0;
#if USE_ASYNC_LDS
                async_copy_b128(gp, lp);
#else
                *(float4*)lp = *(const float4*)gp;
#endif
            } else {
                *(float4*)lp = make_float4(0.f, 0.f, 0.f, 0.f);
            }
        }
#if USE_ASYNC_LDS
        __builtin_amdgcn_s_wait_asynccnt(0);
#endif
        __syncthreads();

        #pragma unroll
        for (int ks = 0; ks < 16; ks += 4) {
            // A fragment: M = l16, K = ks + 2*half + {0,1}  (8B-aligned b64)
            const v2f a = *(const v2f*)&Xs[(wv * 16 + l16) * XS_STRIDE + ks + half * 2];
            #pragma unroll
            for (int t = 0; t < 8; ++t) {
                // B fragment: N = t*16 + l16, K = ks + 2*half + {0,1}
                v2f b;
                const int brow = ks + half * 2;
                const int bcol = t * 16 + l16;
                b.x = Ws[brow * WS_STRIDE + bcol];
                b.y = Ws[(brow + 1) * WS_STRIDE + bcol];
                acc[t] = __builtin_amdgcn_wmma_f32_16x16x4_f32(
                    false, a, false, b, (short)0, acc[t], false, false);
            }
        }
        __syncthreads();
    }

    // epilogue: C/D layout -> row = bm + wv*16 + half*8 + r, col = bn + t*16 + l16
    const int row0 = bm + wv * 16 + half * 8;
    #pragma unroll
    for (int t = 0; t < 8; ++t) {
        const int col = bn + t * 16 + l16;
        if (col >= N) continue;
        const float bv = bias[col];
        #pragma unroll
        for (int r = 0; r < 8; ++r) {
            float v = acc[t][r] + bv;
            if (relu) v = fmaxf(v, 0.f);
            Y[(size_t)(row0 + r) * N + col] = v;
        }
    }
}

// ---------------------------------------------------------------------------
// codebook squared norms: enorm[n] = sum_c E[n][c]^2
// ---------------------------------------------------------------------------
__global__ __launch_bounds__(256)
void enorm_kernel(const float* __restrict__ E, float* __restrict__ enorm)
{
    int n = blockIdx.x * 256 + threadIdx.x;
    if (n < NUM_EMB) {
        float s = 0.f;
        #pragma unroll
        for (int c = 0; c < CODE_DIM; ++c) {
            float v = E[(size_t)n * CODE_DIM + c];
            s += v * v;
        }
        enorm[n] = s;
    }
}

// ---------------------------------------------------------------------------
// Fused VQ: for each of 32768 rows of flat_z, compute argmin_n ||x-e_n||^2
// via metric = ||e||^2 - 2 x.e  (||x||^2 constant per row), then gather E[idx]
// into the quantized output. Block owns 16 rows; 8 waves sweep 256 N-tiles
// with wmma_f32_16x16x4 (K=16 -> 4 wmma per tile).
// ---------------------------------------------------------------------------
__global__ __launch_bounds__(256)
void vq_argmin_gather(const float* __restrict__ Z,      // [32768,16] = enc_z flat
                      const float* __restrict__ E,      // [4096,16]
                      const float* __restrict__ enorm,  // [4096]
                      float* __restrict__ Q)            // [32768,16] quantized
{
    const int tid  = threadIdx.x;
    const int wv   = tid >> 5;
    const int lane = tid & 31;
    const int half = lane >> 4;
    const int l16  = lane & 15;
    const int bm   = blockIdx.x * 16;   // first row of this block

    // A fragments for the 16x16 Z tile (each wave loads its own copy, b64 loads)
    v2f a[4];
    #pragma unroll
    for (int ks = 0; ks < 4; ++ks)
        a[ks] = *(const v2f*)(Z + (size_t)(bm + l16) * CODE_DIM + ks * 4 + half * 2);

    float mval[8];
    int   midx[8];
    #pragma unroll
    for (int r = 0; r < 8; ++r) { mval[r] = 3.4e38f; midx[r] = 0; }

    for (int nt = wv; nt < NUM_EMB / 16; nt += 8) {
        const int n0 = nt * 16;
        v8f acc = {};
        #pragma unroll
        for (int ks = 0; ks < 4; ++ks) {
            // B fragment = E^T tile: B[k][n] = E[n0+n][k]  (b64 load from E row)
            const v2f b = *(const v2f*)(E + (size_t)(n0 + l16) * CODE_DIM + ks * 4 + half * 2);
            acc = __builtin_amdgcn_wmma_f32_16x16x4_f32(
                false, a[ks], false, b, (short)0, acc, false, false);
        }
        const int   n  = n0 + l16;
        const float en = enorm[n];
        #pragma unroll
        for (int r = 0; r < 8; ++r) {
            const float m = en - 2.f * acc[r];  // candidate distance (shifted)
            if (m < mval[r] || (m == mval[r] && n < midx[r])) {
                mval[r] = m; midx[r] = n;
            }
        }
    }

    // reduce across the 16 lanes of each half (xor masks < 16 stay in-half)
    #pragma unroll
    for (int mask = 1; mask < 16; mask <<= 1) {
        #pragma unroll
        for (int r = 0; r < 8; ++r) {
            const float ov = __shfl_xor(mval[r], mask, 32);
            const int   oi = __shfl_xor(midx[r], mask, 32);
            if (ov < mval[r] || (ov == mval[r] && oi < midx[r])) {
                mval[r] = ov; midx[r] = oi;
            }
        }
    }

    __shared__ float sval[8][16];
    __shared__ int   sidx[8][16];
    if (l16 == 0) {
        #pragma unroll
        for (int r = 0; r < 8; ++r) {
            sval[wv][half * 8 + r] = mval[r];
            sidx[wv][half * 8 + r] = midx[r];
        }
    }
    __syncthreads();

    if (tid < 16) {   // one thread per row: combine 8 waves, then gather
        float bv = sval[0][tid];
        int   bi = sidx[0][tid];
        #pragma unroll
        for (int w = 1; w < 8; ++w) {
            const float v = sval[w][tid];
            const int   i = sidx[w][tid];
            if (v < bv || (v == bv && i < bi)) { bv = v; bi = i; }
        }
        #pragma unroll
        for (int c = 0; c < CODE_DIM; ++c)
            Q[(size_t)(bm + tid) * CODE_DIM + c] = E[(size_t)bi * CODE_DIM + c];
    }
}

// ---------------------------------------------------------------------------
// concat kernels
// ---------------------------------------------------------------------------
__global__ __launch_bounds__(256)
void concat_prop_future(const float* __restrict__ prop,
                        const float* __restrict__ fut,
                        float* __restrict__ X)   // [B, 384]
{
    int i = blockIdx.x * 256 + threadIdx.x;
    if (i < B_ROWS * (PROP_DIM + FUT_DIM)) {
        int r = i / (PROP_DIM + FUT_DIM);
        int c = i - r * (PROP_DIM + FUT_DIM);
        X[i] = (c < PROP_DIM) ? prop[(size_t)r * PROP_DIM + c]
                              : fut[(size_t)r * FUT_DIM + (c - PROP_DIM)];
    }
}

__global__ __launch_bounds__(256)
void concat_prop_z(const float* __restrict__ prop,
                   const float* __restrict__ q,   // quantized [B,64]
                   float* __restrict__ X)         // [B, 192]
{
    int i = blockIdx.x * 256 + threadIdx.x;
    if (i < B_ROWS * (PROP_DIM + Z_LEN)) {
        int r = i / (PROP_DIM + Z_LEN);
        int c = i - r * (PROP_DIM + Z_LEN);
        X[i] = (c < PROP_DIM) ? prop[(size_t)r * PROP_DIM + c]
                              : q[(size_t)r * Z_LEN + (c - PROP_DIM)];
    }
}

// ---------------------------------------------------------------------------
// launch
// ---------------------------------------------------------------------------
static inline void launch_gemm(const float* X, const float* W, const float* b,
                               float* Y, int M, int K, int N, int relu,
                               hipStream_t stream)
{
    dim3 grid((N + 127) / 128, M / 128);
    gemm_bias_act<<<grid, 256, 0, stream>>>(X, W, b, Y, M, K, N, relu);
}

extern "C" void kernel_launch(void* const* d_in, const int* in_sizes, int n_in,
                              void* d_out, int out_size, void* d_ws, size_t ws_size,
                              hipStream_t stream)
{
    // inputs in setup_inputs() dict order: prop, future, params{enc, llc, codebook}
    const float* prop   = (const float*)d_in[0];
    const float* future = (const float*)d_in[1];
    const float* eW1 = (const float*)d_in[2];  const float* eb1 = (const float*)d_in[3];
    const float* eW2 = (const float*)d_in[4];  const float* eb2 = (const float*)d_in[5];
    const float* eW3 = (const float*)d_in[6];  const float* eb3 = (const float*)d_in[7];
    const float* eW4 = (const float*)d_in[8];  const float* eb4 = (const float*)d_in[9];
    const float* lW1 = (const float*)d_in[10]; const float* lb1 = (const float*)d_in[11];
    const float* lW2 = (const float*)d_in[12]; const float* lb2 = (const float*)d_in[13];
    const float* lW3 = (const float*)d_in[14]; const float* lb3 = (const float*)d_in[15];
    const float* lW4 = (const float*)d_in[16]; const float* lb4 = (const float*)d_in[17];
    const float* codebook = (const float*)d_in[18];

    // outputs (concatenated flat, return order): out, quantized, enc_z
    float* out_act = (float*)d_out;                                   // [8192,32]
    float* out_q   = out_act + (size_t)B_ROWS * ACT_DIM;              // [8192,64]
    float* out_z   = out_q   + (size_t)B_ROWS * Z_LEN;                // [8192,64]

    // workspace: two 32MB ping-pong activation buffers + e-norms
    float* bufA  = (float*)d_ws;                                      // 8192*1024 f32
    float* bufB  = bufA + (size_t)B_ROWS * 1024;                      // 8192*1024 f32
    float* enorm = bufB + (size_t)B_ROWS * 1024;                      // 4096 f32

    const int nPF = B_ROWS * (PROP_DIM + FUT_DIM);
    const int nPZ = B_ROWS * (PROP_DIM + Z_LEN);

    // ---- encoder ----
    concat_prop_future<<<(nPF + 255) / 256, 256, 0, stream>>>(prop, future, bufA);
    launch_gemm(bufA, eW1, eb1, bufB, B_ROWS, 384, 1024, 1, stream);
    launch_gemm(bufB, eW2, eb2, bufA, B_ROWS, 1024, 1024, 1, stream);
    launch_gemm(bufA, eW3, eb3, bufB, B_ROWS, 1024, 512, 1, stream);
    launch_gemm(bufB, eW4, eb4, out_z, B_ROWS, 512, Z_LEN, 0, stream);   // enc_z

    // ---- vector quantization ----
    enorm_kernel<<<(NUM_EMB + 255) / 256, 256, 0, stream>>>(codebook, enorm);
    vq_argmin_gather<<<(B_ROWS * 4) / 16, 256, 0, stream>>>(out_z, codebook, enorm, out_q);

    // ---- LLC (z_curr == quantized in forward value) ----
    concat_prop_z<<<(nPZ + 255) / 256, 256, 0, stream>>>(prop, out_q, bufA);
    launch_gemm(bufA, lW1, lb1, bufB, B_ROWS, 192, 1024, 1, stream);
    launch_gemm(bufB, lW2, lb2, bufA, B_ROWS, 1024, 1024, 1, stream);
    launch_gemm(bufA, lW3, lb3, bufB, B_ROWS, 1024, 512, 1, stream);
    launch_gemm(bufB, lW4, lb4, out_act, B_ROWS, 512, ACT_DIM, 0, stream);
}